// FarthestPointSampling_20985210208580
// MI455X (gfx1250) — compile-verified
//
#include <hip/hip_runtime.h>
#include <hip/hip_bf16.h>
#include <limits.h>

// FPS: B=32 clouds, N=32768 points (float3), sample NP=1024.
// One 1024-thread workgroup per cloud. Point idx = j*1024 + tid (j = 0..31).
// x,y in LDS (256 KB of the 320 KB/WGP), z + running min-distance in VGPRs.

#define FPS_B   32
#define FPS_N   32768
#define FPS_NP  1024
#define FPS_T   1024
#define FPS_PPT 32  // points per thread = N / T

// Dynamic LDS layout (in floats):
//   [0, 65536)          s_xy  : interleaved (x,y) per point       262144 B
//   [65536, 68608)      s_out : NP*3 sampled points                12288 B
//   [68608, 68640)      s_red : 32 per-wave maxima
//   68640 s_bmax, 68641 s_cz, 68642 s_far(int), 68643 s_win(int)
#define LDS_FLOATS (65536 + 3072 + 36)

extern "C" __global__ void __launch_bounds__(FPS_T, 1)
fps_kernel(const float* __restrict__ in, float* __restrict__ out)
{
    extern __shared__ float smem[];
    float* s_xy   = smem;
    float* s_out  = smem + 65536;
    float* s_red  = smem + 65536 + 3072;
    float* s_bmax = s_red + 32;
    float* s_cz   = s_red + 33;
    int*   s_far  = (int*)(s_red + 34);
    int*   s_win  = (int*)(s_red + 35);

    const int tid = threadIdx.x;
    const int b   = blockIdx.x;
    const float* __restrict__ p = in + (size_t)b * (FPS_N * 3);

#if defined(__gfx1250__)
    __builtin_prefetch(p + (size_t)tid * 3, 0, 0);  // global_prefetch_b8
#endif

    float zreg[FPS_PPT];
    float dist[FPS_PPT];

    // ---- load: x,y -> LDS, z -> VGPRs -------------------------------------
#pragma unroll
    for (int j = 0; j < FPS_PPT; ++j) {
        const int idx = j * FPS_T + tid;
        const float x  = p[idx * 3 + 0];
        const float y  = p[idx * 3 + 1];
        const float zz = p[idx * 3 + 2];
        s_xy[2 * idx + 0] = x;
        s_xy[2 * idx + 1] = y;
        zreg[j] = zz;
        dist[j] = 1e10f;
    }
    if (tid == 0) {
        *s_far = 0;            // first farthest index (reference starts at 0)
        *s_win = INT_MAX;
        *s_cz  = zreg[0];      // point 0 is owned by (tid=0, j=0)
    }
    __syncthreads();

    const int lane = tid & 31;

    // ---- 1024 strictly sequential FPS iterations --------------------------
#pragma unroll 1
    for (int k = 0; k < FPS_NP; ++k) {
        const int   far = *s_far;
        const float cx  = s_xy[2 * far + 0];   // LDS broadcast (same addr)
        const float cy  = s_xy[2 * far + 1];
        const float cz  = *s_cz;

        // emit BEFORE update (matches reference): out[k] = p[farthest_k]
        if (tid == 0) {
            s_out[3 * k + 0] = cx;
            s_out[3 * k + 1] = cy;
            s_out[3 * k + 2] = cz;
        }

        float best = -1.0f;
#pragma unroll
        for (int j = 0; j < FPS_PPT; ++j) {
            const int idx = j * FPS_T + tid;
            const float x  = s_xy[2 * idx + 0];
            const float y  = s_xy[2 * idx + 1];
            const float dx = x - cx;
            const float dy = y - cy;
            const float dz = zreg[j] - cz;
            const float d  = fmaf(dz, dz, fmaf(dy, dy, dx * dx));
            const float nd = fminf(dist[j], d);
            dist[j] = nd;
            best = fmaxf(best, nd);
        }

        // wave32 max reduction
#pragma unroll
        for (int m = 16; m >= 1; m >>= 1)
            best = fmaxf(best, __shfl_xor(best, m, 32));
        if (lane == 0) s_red[tid >> 5] = best;
        __syncthreads();                                   // B1

        if (tid < 32) {
            float v = s_red[tid];
#pragma unroll
            for (int m = 16; m >= 1; m >>= 1)
                v = fmaxf(v, __shfl_xor(v, m, 32));
            if (tid == 0) *s_bmax = v;
        } else if (tid == 32) {
            *s_win = INT_MAX;   // reset winner slot for this iteration
        }
        __syncthreads();                                   // B2

        // exact argmax with first-occurrence tie-break: ds atomic min on the
        // true flat point index (shuffle max is bit-exact, so == is safe).
        const float bmax = *s_bmax;
#pragma unroll
        for (int j = 0; j < FPS_PPT; ++j) {
            if (dist[j] == bmax) atomicMin(s_win, j * FPS_T + tid);
        }
        __syncthreads();                                   // B3

        const int win = *s_win;
        // winner republishes its z (static register indexing via unroll)
#pragma unroll
        for (int j = 0; j < FPS_PPT; ++j) {
            if (win == j * FPS_T + tid) *s_cz = zreg[j];
        }
        if (tid == 0) *s_far = win;
        __syncthreads();                                   // B4
    }

    // ---- flush sampled points: LDS -> global ------------------------------
    float* __restrict__ dst = out + (size_t)b * (FPS_NP * 3);
#if defined(__gfx1250__) && \
    __has_builtin(__builtin_amdgcn_global_store_async_from_lds_b128) && \
    __has_builtin(__builtin_amdgcn_s_wait_asynccnt)
    typedef int v4i __attribute__((vector_size(16)));
    typedef __attribute__((address_space(1))) v4i* gv4i_ptr;  // global (AS1)
    typedef __attribute__((address_space(3))) v4i* lv4i_ptr;  // LDS    (AS3)
    if (tid < (FPS_NP * 3) / 4) {   // 768 lanes x 16B = 12288 B
        __builtin_amdgcn_global_store_async_from_lds_b128(
            (gv4i_ptr)(dst + tid * 4), (lv4i_ptr)(s_out + tid * 4), 0, 0);
    }
    __builtin_amdgcn_s_wait_asynccnt(0);
#else
    for (int i = tid; i < FPS_NP * 3; i += FPS_T) dst[i] = s_out[i];
#endif
}

extern "C" void kernel_launch(void* const* d_in, const int* in_sizes, int n_in,
                              void* d_out, int out_size, void* d_ws, size_t ws_size,
                              hipStream_t stream)
{
    (void)in_sizes; (void)n_in; (void)out_size; (void)d_ws; (void)ws_size;
    const float* in  = (const float*)d_in[0];
    float*       out = (float*)d_out;
    const size_t shmem = (size_t)LDS_FLOATS * sizeof(float);  // 274,576 B
    fps_kernel<<<dim3(FPS_B), dim3(FPS_T), shmem, stream>>>(in, out);
}